// PointNetFeaturePropagation_75136157876260
// MI455X (gfx1250) — compile-verified
//
#include <hip/hip_runtime.h>
#include <hip/hip_bf16.h>

// PointNetFeaturePropagation for MI455X (gfx1250, wave32).
// Pipeline:
//   K1: per-block (256 queries): 3-NN scan over LDS-cached xyz2, IDW interp of
//       points2 into LDS, then fp32 WMMA GEMM (interp @ W^T + b) -> d_out,
//       accumulating per-channel sum / sumsq via global atomics.
//   K2: finalize BN stats (mean, rstd) for 64 channels.
//   K3: elementwise BN apply in-place on d_out (float4 vectorized).

typedef float v2f __attribute__((ext_vector_type(2)));
typedef float v8f __attribute__((ext_vector_type(8)));

#define BB      4
#define NN      16384
#define SS      4096
#define DD      64
#define TPB     256   // threads per block (8 wave32)
#define WPPITCH 65    // W-pair LDS pitch (float2 units) per k-pair row
#define IPITCH  68    // interp LDS row pitch (floats, %4==0 -> 16B-aligned rows)

// LDS float totals
#define XYZ2_F  (SS * 4)                  // 16384 floats (float4 padded xyz2)
#define WP_F    ((DD / 2) * WPPITCH * 2)  // 32 pair-rows * 65 float2 = 4160 floats
#define INT_F   (TPB * IPITCH)            // 17408 floats
#define SMEM_F  (XYZ2_F + WP_F + INT_F)

__global__ __launch_bounds__(TPB)
void fp_knn_interp_gemm_kernel(const float* __restrict__ xyz1,
                               const float* __restrict__ xyz2,
                               const float* __restrict__ points2,
                               const float* __restrict__ W,
                               const float* __restrict__ bias,
                               float* __restrict__ out,
                               float* __restrict__ gsum,    // [64]
                               float* __restrict__ gsumsq)  // [64]
{
    extern __shared__ float smem[];
    float4* s_xyz2 = (float4*)smem;                 // 4096 float4 = 64 KB
    float*  s_WpF  = smem + XYZ2_F;                 // pair-interleaved W^T
    v2f*    s_Wp   = (v2f*)s_WpF;                   // [32 pair-rows][WPPITCH]
    float*  s_int  = s_WpF + WP_F;                  // [256][IPITCH]

    const int t   = threadIdx.x;
    const int bpb = NN / TPB;                       // 64 blocks per batch
    const int b   = blockIdx.x / bpb;
    const int n0  = (blockIdx.x % bpb) * TPB;

    // ---- stage xyz2[b] (padded float4) and pair-interleaved W^T into LDS ----
    const float* x2 = xyz2 + (size_t)b * SS * 3;
    for (int j = t; j < SS; j += TPB) {
        s_xyz2[j] = make_float4(x2[j * 3 + 0], x2[j * 3 + 1], x2[j * 3 + 2], 0.0f);
    }
    for (int i = t; i < DD * DD; i += TPB) {
        int e = i >> 6, d = i & 63;
        // s_Wp[d/2][e] = { W[e][d_even], W[e][d_even+1] }
        s_WpF[((d >> 1) * WPPITCH + e) * 2 + (d & 1)] = W[i];  // W[e][d] at e*64+d
    }
    __syncthreads();

    // ---- per-thread 3-NN scan over S=4096 candidates (LDS broadcast reads) ----
    const int n = n0 + t;
    const float* q = xyz1 + ((size_t)b * NN + n) * 3;
    const float qx = q[0], qy = q[1], qz = q[2];

    float d0 = 3.0e38f, d1 = 3.0e38f, d2 = 3.0e38f;
    int   i0 = 0, i1 = 0, i2 = 0;
#pragma unroll 4
    for (int j = 0; j < SS; ++j) {
        float4 p = s_xyz2[j];
        float dx = qx - p.x, dy = qy - p.y, dz = qz - p.z;
        float dd = dx * dx + dy * dy + dz * dz;
        if (dd < d2) {
            if (dd < d1) {
                d2 = d1; i2 = i1;
                if (dd < d0) { d1 = d0; i1 = i0; d0 = dd; i0 = j; }
                else         { d1 = dd; i1 = j; }
            } else { d2 = dd; i2 = j; }
        }
    }

    // ---- inverse-distance weights + gather/interp into LDS ----
    const float r0 = 1.0f / (d0 + 1e-8f);
    const float r1 = 1.0f / (d1 + 1e-8f);
    const float r2 = 1.0f / (d2 + 1e-8f);
    const float rs = 1.0f / (r0 + r1 + r2);
    const float w0 = r0 * rs, w1 = r1 * rs, w2 = r2 * rs;

    const float* P   = points2 + (size_t)b * SS * DD;
    const float4* p0 = (const float4*)(P + (size_t)i0 * DD);
    const float4* p1 = (const float4*)(P + (size_t)i1 * DD);
    const float4* p2 = (const float4*)(P + (size_t)i2 * DD);
#pragma unroll
    for (int c4 = 0; c4 < DD / 4; ++c4) {
        float4 a = p0[c4], bb = p1[c4], cc = p2[c4];
        float4 r;
        r.x = w0 * a.x + w1 * bb.x + w2 * cc.x;
        r.y = w0 * a.y + w1 * bb.y + w2 * cc.y;
        r.z = w0 * a.z + w1 * bb.z + w2 * cc.z;
        r.w = w0 * a.w + w1 * bb.w + w2 * cc.w;
        *(float4*)(s_int + t * IPITCH + c4 * 4) = r;   // 16B-aligned (IPITCH%4==0)
    }
    __syncthreads();

    // ---- fp32 WMMA GEMM: out[256,64] = interp[256,64] @ W^T, + bias ----
    // wave w owns row tiles {2w, 2w+1}, all 4 col tiles. 16x16x4 fragments:
    //   A: lanes 0-15 -> M=lane, K={k0,k0+1}; lanes 16-31 -> M=lane-16, K={k0+2,k0+3}
    //   B: lanes 0-15 -> K={k0,k0+1}, N=lane; lanes 16-31 -> K={k0+2,k0+3}, N=lane-16
    //   C/D: lanes 0-15 -> N=lane, M=r; lanes 16-31 -> N=lane-16, M=8+r
    const int wave  = t >> 5;
    const int lane  = t & 31;
    const int lhalf = lane >> 4;
    const int lmod  = lane & 15;

    for (int nt = 0; nt < 4; ++nt) {
        const int ecol = nt * 16 + lmod;
        const float bv = bias[ecol];
        float s = 0.0f, s2 = 0.0f;
        for (int mt = 0; mt < 2; ++mt) {
            const int row0 = (wave * 2 + mt) * 16;
            v8f acc = {0.f, 0.f, 0.f, 0.f, 0.f, 0.f, 0.f, 0.f};
            const float* arow = s_int + (row0 + lmod) * IPITCH;
#pragma unroll
            for (int k0 = 0; k0 < DD; k0 += 4) {
                const int ka = k0 + 2 * lhalf;                       // even
                v2f A  = *(const v2f*)(arow + ka);                   // ds_load_b64
                v2f Bf = s_Wp[(ka >> 1) * WPPITCH + ecol];           // ds_load_b64
                acc = __builtin_amdgcn_wmma_f32_16x16x4_f32(
                    false, A, false, Bf, (short)0, acc, false, false);
            }
            float* orow = out + (((size_t)b * NN) + n0 + row0 + lhalf * 8) * DD + ecol;
#pragma unroll
            for (int r = 0; r < 8; ++r) {
                float v = acc[r] + bv;
                orow[(size_t)r * DD] = v;
                s  += v;
                s2 += v * v;
            }
        }
        // lanes l and l^16 hold the same channel -> combine, then 16 atomics/wave
        s  += __shfl_xor(s, 16, 32);
        s2 += __shfl_xor(s2, 16, 32);
        if (lhalf == 0) {
            atomicAdd(&gsum[ecol], s);
            atomicAdd(&gsumsq[ecol], s2);
        }
    }
}

__global__ void fp_bn_stats_kernel(const float* __restrict__ gsum,
                                   const float* __restrict__ gsumsq,
                                   float* __restrict__ mean,
                                   float* __restrict__ rstd)
{
    const int c = threadIdx.x;
    const float inv_n = 1.0f / (float)(BB * NN);
    const float m = gsum[c] * inv_n;
    const float v = gsumsq[c] * inv_n - m * m;
    mean[c] = m;
    rstd[c] = rsqrtf(v + 1e-5f);
}

__global__ __launch_bounds__(TPB)
void fp_bn_apply_kernel(float* __restrict__ out,
                        const float* __restrict__ mean,
                        const float* __restrict__ rstd,
                        const float* __restrict__ gamma,
                        const float* __restrict__ beta)
{
    const size_t i4 = (size_t)blockIdx.x * TPB + threadIdx.x;   // float4 index
    const int c = (int)((i4 * 4) & (DD - 1));                   // base channel (%4==0)
    float4 v = ((const float4*)out)[i4];
    v.x = gamma[c + 0] * (v.x - mean[c + 0]) * rstd[c + 0] + beta[c + 0];
    v.y = gamma[c + 1] * (v.y - mean[c + 1]) * rstd[c + 1] + beta[c + 1];
    v.z = gamma[c + 2] * (v.z - mean[c + 2]) * rstd[c + 2] + beta[c + 2];
    v.w = gamma[c + 3] * (v.w - mean[c + 3]) * rstd[c + 3] + beta[c + 3];
    ((float4*)out)[i4] = v;
}

extern "C" void kernel_launch(void* const* d_in, const int* in_sizes, int n_in,
                              void* d_out, int out_size, void* d_ws, size_t ws_size,
                              hipStream_t stream) {
    const float* xyz1    = (const float*)d_in[0];
    const float* xyz2    = (const float*)d_in[1];
    // d_in[2] = points1 (unused in this module variant)
    const float* points2 = (const float*)d_in[3];
    const float* W       = (const float*)d_in[4];
    const float* bias    = (const float*)d_in[5];
    const float* gamma   = (const float*)d_in[6];
    const float* beta    = (const float*)d_in[7];
    float* out = (float*)d_out;

    float* gsum   = (float*)d_ws;        // [64]
    float* gsumsq = gsum + 64;           // [64]
    float* mean   = gsum + 128;          // [64]
    float* rstd   = gsum + 192;          // [64]

    hipMemsetAsync(d_ws, 0, 256 * sizeof(float), stream);

    const size_t shmem = (size_t)SMEM_F * sizeof(float);
    fp_knn_interp_gemm_kernel<<<BB * (NN / TPB), TPB, shmem, stream>>>(
        xyz1, xyz2, points2, W, bias, out, gsum, gsumsq);

    fp_bn_stats_kernel<<<1, DD, 0, stream>>>(gsum, gsumsq, mean, rstd);

    fp_bn_apply_kernel<<<(BB * NN * DD) / (TPB * 4), TPB, 0, stream>>>(
        out, mean, rstd, gamma, beta);
}